// MMDNormalLoss_61727269978470
// MI455X (gfx1250) — compile-verified
//
#include <hip/hip_runtime.h>
#include <hip/hip_bf16.h>

typedef __attribute__((ext_vector_type(2))) float v2f;
typedef __attribute__((ext_vector_type(8))) float v8f;

#define N_PTS 8192
#define DIM   64
#define BM    128            // tile rows  (grid.y)
#define BN    64             // tile cols  (grid.x)
#define LDS_STRIDE 68        // floats: rows 16B-aligned, bank-conflict-free wmma reads

// workspace layout (in floats)
#define WS_S    0
#define WS_SN   (N_PTS * DIM)            // 524288
#define WS_ZN   (WS_SN + N_PTS)
#define WS_PART (WS_ZN + N_PTS)
#define NBLK    ((N_PTS / BN) * (N_PTS / BM) * 3)   // 128*64*3 = 24576

// ---------------------------------------------------------------------------
// Kernel 1: samples = exp(log_sigma)*eps + mu ; row squared-norms of S and Z.
// One wave32 per row (each lane owns columns lane and lane+32).
// ---------------------------------------------------------------------------
__global__ __launch_bounds__(256)
void prep_kernel(const float* __restrict__ z, const float* __restrict__ eps,
                 const float* __restrict__ mu, const float* __restrict__ log_sigma,
                 float* __restrict__ S, float* __restrict__ sn, float* __restrict__ zn) {
    int wave = threadIdx.x >> 5;
    int lane = threadIdx.x & 31;
    int row  = blockIdx.x * 8 + wave;
    if (row >= N_PTS) return;

    float sigma = __expf(log_sigma[0]);
    float m     = mu[0];

    size_t base = (size_t)row * DIM;
    float e0 = eps[base + lane];
    float e1 = eps[base + lane + 32];
    float s0 = sigma * e0 + m;
    float s1 = sigma * e1 + m;
    S[base + lane]      = s0;
    S[base + lane + 32] = s1;

    float z0 = z[base + lane];
    float z1 = z[base + lane + 32];

    float accs = s0 * s0 + s1 * s1;
    float accz = z0 * z0 + z1 * z1;
    #pragma unroll
    for (int off = 16; off; off >>= 1) {
        accs += __shfl_xor(accs, off, 32);
        accz += __shfl_xor(accz, off, 32);
    }
    if (lane == 0) { sn[row] = accs; zn[row] = accz; }
}

// ---------------------------------------------------------------------------
// Kernel 2: one 128x64 tile of exp(-(x2 + y2 - 2 x.y)/D^2), summed.
// grid.z selects (X,Y,weight): 0 -> (S,S,+1), 1 -> (Z,Z,+1), 2 -> (S,Z,-2).
// 8 waves; wave w computes rows [16w,16w+16) x all 64 cols as four 16x16
// WMMA tiles (16 x v_wmma_f32_16x16x4_f32 each, K=64).
// ---------------------------------------------------------------------------
__global__ __launch_bounds__(256)
void mmd_tile_kernel(const float* __restrict__ S, const float* __restrict__ Z,
                     const float* __restrict__ sn, const float* __restrict__ zn,
                     float* __restrict__ partials) {
    __shared__ float Xs[BM * LDS_STRIDE];
    __shared__ float Ys[BN * LDS_STRIDE];
    __shared__ float red[8];

    const float *X, *Y, *xn, *yn;
    float wgt;
    if (blockIdx.z == 0)      { X = S; Y = S; xn = sn; yn = sn; wgt =  1.0f; }
    else if (blockIdx.z == 1) { X = Z; Y = Z; xn = zn; yn = zn; wgt =  1.0f; }
    else                      { X = S; Y = Z; xn = sn; yn = zn; wgt = -2.0f; }

    const int rowBase = blockIdx.y * BM;
    const int colBase = blockIdx.x * BN;
    const int t = threadIdx.x;

    // Cooperative tile staging, float4 per transaction (global 16B aligned,
    // LDS rows 16B aligned since LDS_STRIDE*4 = 272 is a multiple of 16).
    #pragma unroll
    for (int i = 0; i < 8; ++i) {               // X: 128x64 = 2048 float4
        int f  = t + i * 256;
        int r  = f >> 4;                        // 16 float4 per row
        int c4 = f & 15;
        float4 v = ((const float4*)(X + (size_t)(rowBase + r) * DIM))[c4];
        *(float4*)&Xs[r * LDS_STRIDE + c4 * 4] = v;
    }
    #pragma unroll
    for (int i = 0; i < 4; ++i) {               // Y: 64x64 = 1024 float4
        int f  = t + i * 256;
        int r  = f >> 4;
        int c4 = f & 15;
        float4 v = ((const float4*)(Y + (size_t)(colBase + r) * DIM))[c4];
        *(float4*)&Ys[r * LDS_STRIDE + c4 * 4] = v;
    }
    __syncthreads();

    const int wave   = t >> 5;
    const int lane   = t & 31;
    const int lane16 = lane & 15;
    const int half   = lane >> 4;       // 0: K+{0,1}, 1: K+{2,3} (ISA 7.12.2)
    const int m0     = wave * 16;

    // A fragment row for this lane (A 16x4 f32 layout: lanes 0-15 M=lane K=k,k+1;
    // lanes 16-31 M=lane-16 K=k+2,k+3). B (4x16) addresses identically for
    // row-major Y since B[k][n] = Y[n][k].
    const float* xrow = &Xs[(m0 + lane16) * LDS_STRIDE + 2 * half];

    const float inv = 1.0f / (float)(DIM * DIM);   // 1/4096
    float lsum = 0.0f;

    #pragma unroll
    for (int tc = 0; tc < BN / 16; ++tc) {
        const float* yrow = &Ys[(tc * 16 + lane16) * LDS_STRIDE + 2 * half];
        v8f acc = {};
        #pragma unroll
        for (int kk = 0; kk < DIM; kk += 4) {
            v2f a = *(const v2f*)(xrow + kk);
            v2f b = *(const v2f*)(yrow + kk);
            acc = __builtin_amdgcn_wmma_f32_16x16x4_f32(
                      false, a, false, b, (short)0, acc, false, false);
        }
        // C/D layout: VGPR g holds (M = g + 8*half, N = lane16).
        float ynv = yn[colBase + tc * 16 + lane16];
        int   rb  = rowBase + m0 + 8 * half;
        #pragma unroll
        for (int g = 0; g < 8; ++g) {
            float xnv = xn[rb + g];
            lsum += __expf((2.0f * acc[g] - xnv - ynv) * inv);
        }
    }

    // Block reduction (fixed shape -> deterministic).
    #pragma unroll
    for (int off = 16; off; off >>= 1) lsum += __shfl_xor(lsum, off, 32);
    if (lane == 0) red[wave] = lsum;
    __syncthreads();
    if (t == 0) {
        float s = 0.0f;
        #pragma unroll
        for (int w = 0; w < 8; ++w) s += red[w];
        partials[((size_t)blockIdx.z * gridDim.y + blockIdx.y) * gridDim.x
                 + blockIdx.x] = s * wgt;
    }
}

// ---------------------------------------------------------------------------
// Kernel 3: deterministic reduction of per-block partials -> scalar output.
// ---------------------------------------------------------------------------
__global__ __launch_bounds__(256)
void finalize_kernel(const float* __restrict__ partials, float* __restrict__ out) {
    __shared__ float red[256];
    float s = 0.0f;
    for (int i = threadIdx.x; i < NBLK; i += 256) s += partials[i];
    red[threadIdx.x] = s;
    __syncthreads();
    for (int off = 128; off; off >>= 1) {
        if (threadIdx.x < off) red[threadIdx.x] += red[threadIdx.x + off];
        __syncthreads();
    }
    if (threadIdx.x == 0)
        out[0] = red[0] * (1.0f / ((float)N_PTS * (float)N_PTS));
}

extern "C" void kernel_launch(void* const* d_in, const int* in_sizes, int n_in,
                              void* d_out, int out_size, void* d_ws, size_t ws_size,
                              hipStream_t stream) {
    const float* z   = (const float*)d_in[0];
    const float* eps = (const float*)d_in[1];
    const float* mu  = (const float*)d_in[2];
    const float* ls  = (const float*)d_in[3];

    float* ws       = (float*)d_ws;
    float* S        = ws + WS_S;
    float* sn       = ws + WS_SN;
    float* zn       = ws + WS_ZN;
    float* partials = ws + WS_PART;
    float* out      = (float*)d_out;

    prep_kernel<<<N_PTS / 8, 256, 0, stream>>>(z, eps, mu, ls, S, sn, zn);

    dim3 grid(N_PTS / BN, N_PTS / BM, 3);   // 128 x 64 x 3
    mmd_tile_kernel<<<grid, 256, 0, stream>>>(S, z, sn, zn, partials);

    finalize_kernel<<<1, 256, 0, stream>>>(partials, out);
}